// GIN_43404939494170
// MI455X (gfx1250) — compile-verified
//
#include <hip/hip_runtime.h>
#include <hip/hip_bf16.h>

typedef __attribute__((ext_vector_type(2))) float v2f;
typedef __attribute__((ext_vector_type(8))) float v8f;

#define G_TOT 16384
#define N_NODES 32
#define D_IN 128
#define H_HID 256
#define R_RANK 32
#define O_OUT 64
#define WAVES_PER_BLOCK 8

// ---------------------------------------------------------------------------
// Kernel 1: U = Wl @ V  -> [64, 32].  Tiny (1 MFLOP), runs once into d_ws.
// score = pooled @ U^T + bl  replaces (pooled @ V^T) @ Wl^T.
// ---------------------------------------------------------------------------
__global__ void gin_precompute_U(const float* __restrict__ Wl,
                                 const float* __restrict__ V,
                                 float* __restrict__ U) {
    int t = blockIdx.x * blockDim.x + threadIdx.x;   // 0..2047
    if (t >= O_OUT * R_RANK) return;
    int o = t >> 5;          // output row 0..63
    int r = t & 31;          // rank     0..31
    float s = 0.0f;
    #pragma unroll 8
    for (int hh = 0; hh < H_HID; ++hh)
        s = fmaf(Wl[o * H_HID + hh], V[hh * R_RANK + r], s);
    U[t] = s;
}

// ---------------------------------------------------------------------------
// Kernel 2: main.  One graph per wave; 8 waves (256 thr) per block.
//   feat = h_g @ W via v_wmma_f32_16x16x4_f32 (exact fp32),
//   node-product reduction over M (across C VGPRs + shfl_xor 16),
//   score = pooled @ U^T + bl.
// ---------------------------------------------------------------------------
__global__ void __launch_bounds__(256)
gin_main(const float* __restrict__ h,
         const float* __restrict__ W,
         const float* __restrict__ U,
         const float* __restrict__ bl,
         float* __restrict__ out) {
    // W swizzled to per-lane WMMA B layout: [k=0..31][ntile=0..1][lane=0..31]
    __shared__ v2f   ldsW[32 * 2 * 32];          // 16 KB
    __shared__ float ldsU[O_OUT * 33];           // 64 rows padded to 33 (8.4 KB)
    __shared__ float ldsPool[WAVES_PER_BLOCK * R_RANK];

    const int tid = threadIdx.x;

    // ---- stage W into LDS in WMMA B-operand layout --------------------------
    // B (KxN=4x16) layout: lane<16 holds N=lane, (vgpr0,vgpr1) = (K0,K1);
    //                      lane>=16 holds N=lane-16, (K2,K3).
    for (int i = tid; i < 32 * 2 * 32; i += 256) {
        int k    = i >> 6;
        int nt   = (i >> 5) & 1;
        int lane = i & 31;
        int hi   = lane >> 4;
        int col  = 16 * nt + (lane & 15);
        int row  = 4 * k + 2 * hi;
        v2f v;
        v.x = W[row * R_RANK + col];
        v.y = W[(row + 1) * R_RANK + col];
        ldsW[i] = v;
    }
    // ---- stage U into LDS, padded rows --------------------------------------
    for (int i = tid; i < O_OUT * R_RANK; i += 256)
        ldsU[(i >> 5) * 33 + (i & 31)] = U[i];
    __syncthreads();

    const int wave = tid >> 5;
    const int lane = tid & 31;
    const int hi   = lane >> 4;           // 0 | 1
    const int lo   = lane & 15;
    const int g    = blockIdx.x * WAVES_PER_BLOCK + wave;

    // A (MxK=16x4) layout: lane<16 holds M=lane, (vgpr0,vgpr1)=(K0,K1);
    //                      lane>=16 holds M=lane-16, (K2,K3).
    const float* hg  = h + (size_t)g * (N_NODES * D_IN);
    const float* a0p = hg + lo * D_IN + 2 * hi;            // M-tile 0 (nodes 0-15)
    const float* a1p = a0p + 16 * D_IN;                    // M-tile 1 (nodes 16-31)

    v8f acc00 = {}; v8f acc01 = {}; v8f acc10 = {}; v8f acc11 = {};

    #pragma unroll 4
    for (int k = 0; k < 32; ++k) {
        v2f a0 = *(const v2f*)(a0p + 4 * k);
        v2f a1 = *(const v2f*)(a1p + 4 * k);
        v2f b0 = ldsW[(k * 2 + 0) * 32 + lane];
        v2f b1 = ldsW[(k * 2 + 1) * 32 + lane];
        acc00 = __builtin_amdgcn_wmma_f32_16x16x4_f32(false, a0, false, b0,
                                                      (short)0, acc00, false, false);
        acc01 = __builtin_amdgcn_wmma_f32_16x16x4_f32(false, a0, false, b1,
                                                      (short)0, acc01, false, false);
        acc10 = __builtin_amdgcn_wmma_f32_16x16x4_f32(false, a1, false, b0,
                                                      (short)0, acc10, false, false);
        acc11 = __builtin_amdgcn_wmma_f32_16x16x4_f32(false, a1, false, b1,
                                                      (short)0, acc11, false, false);
    }

    // ---- product over the 32 nodes (M dimension) ----------------------------
    // C layout: VGPR v, lane<16 -> (M=v,     N=lane),
    //                   lane>=16 -> (M=v+8,  N=lane-16).
    float p0 = 1.0f, p1 = 1.0f;
    #pragma unroll
    for (int i = 0; i < 8; ++i) { p0 *= acc00[i]; p1 *= acc01[i]; }
    #pragma unroll
    for (int i = 0; i < 8; ++i) { p0 *= acc10[i]; p1 *= acc11[i]; }
    // lane l and lane l^16 hold complementary M halves of the same column N
    p0 *= __shfl_xor(p0, 16, 32);
    p1 *= __shfl_xor(p1, 16, 32);

    // pooled[r] for this wave's graph (wave-private region, no barrier needed)
    if (lane < 16) {
        ldsPool[wave * R_RANK + lo]      = p0;   // ranks  0..15
        ldsPool[wave * R_RANK + 16 + lo] = p1;   // ranks 16..31
    }

    // ---- score[g][o] = sum_r pooled[r] * U[o][r] + bl[o] --------------------
    float s0 = 0.0f, s1 = 0.0f;
    #pragma unroll
    for (int r = 0; r < R_RANK; ++r) {
        float pv = ldsPool[wave * R_RANK + r];           // broadcast
        s0 = fmaf(pv, ldsU[lane * 33 + r], s0);          // outputs  0..31
        s1 = fmaf(pv, ldsU[(lane + 32) * 33 + r], s1);   // outputs 32..63
    }
    float* og = out + (size_t)g * O_OUT;
    og[lane]      = s0 + bl[lane];
    og[lane + 32] = s1 + bl[lane + 32];
}

// ---------------------------------------------------------------------------
extern "C" void kernel_launch(void* const* d_in, const int* in_sizes, int n_in,
                              void* d_out, int out_size, void* d_ws, size_t ws_size,
                              hipStream_t stream) {
    const float* h  = (const float*)d_in[0];   // [G,N,D]
    const float* W  = (const float*)d_in[1];   // [D,R]
    const float* V  = (const float*)d_in[2];   // [H,R]
    const float* Wl = (const float*)d_in[3];   // [O,H]
    const float* bl = (const float*)d_in[4];   // [O]
    float* out = (float*)d_out;                // [G,O]
    float* U   = (float*)d_ws;                 // [O,R] = 8 KB scratch

    gin_precompute_U<<<(O_OUT * R_RANK + 255) / 256, 256, 0, stream>>>(Wl, V, U);
    gin_main<<<G_TOT / WAVES_PER_BLOCK, 256, 0, stream>>>(h, W, U, bl, out);
}